// DetrLoss_21260088115997
// MI455X (gfx1250) — compile-verified
//
#include <hip/hip_runtime.h>
#include <stdint.h>

#define QN 300
#define BN 256
#define TILE_J 32
#define JT 10            // ceil(300/32)
#define EPSF 1e-7f

typedef unsigned int tdm_v4u __attribute__((ext_vector_type(4)));
typedef int          tdm_v4i __attribute__((ext_vector_type(4)));
typedef int          tdm_v8i __attribute__((ext_vector_type(8)));

__device__ __forceinline__ float wave_max32(float v) {
#pragma unroll
  for (int m = 16; m; m >>= 1) v = fmaxf(v, __shfl_xor(v, m, 32));
  return v;
}
__device__ __forceinline__ float wave_sum32(float v) {
#pragma unroll
  for (int m = 16; m; m >>= 1) v += __shfl_xor(v, m, 32);
  return v;
}

__global__ __launch_bounds__(256) void detr_cost_ce_kernel(
    const float* __restrict__ pred_boxes,
    const float* __restrict__ pred_cls,
    const float* __restrict__ gt_boxes,
    const int*   __restrict__ gt_cls,
    const unsigned char* __restrict__ gt_validity,
    float* __restrict__ cost,
    float* __restrict__ ws)
{
  __shared__ float tile[TILE_J * QN];   // pred_cls[b, j0+r, i], r-major: 38400 B
  __shared__ float sh_ce[8];

  const int jt = blockIdx.x;            // j-tile 0..9
  const int b  = blockIdx.y;            // batch 0..255
  const int j0 = jt * TILE_J;
  const int t  = threadIdx.x;

  // ---- TDM async tensor load: pred_cls[b, j0:j0+32, 0:300] -> LDS ----
  if (t < 32) {                         // wave 0 only (wave32)
    uint64_t gaddr = (uint64_t)(uintptr_t)(pred_cls + ((size_t)b * QN + j0) * QN);
    uint32_t ldsa  = (uint32_t)(uintptr_t)(void*)&tile[0];
    tdm_v4u g0 = { 1u,                                  // count = 1 (valid D#)
                   ldsa,                                // lds_addr
                   (uint32_t)gaddr,                     // global_addr[31:0]
                   ((uint32_t)(gaddr >> 32) & 0x1FFFFFFu) | 0x80000000u }; // addr hi | type=2
    const int dim1 = QN - j0;                           // rows left; OOB rows -> zero fill
    tdm_v8i g1 = { (int)(2u << 16),                     // data_size = 4 bytes
                   (int)(((uint32_t)QN)   << 16),       // tensor_dim0 = 300 (bits 63:48)
                   (int)(((uint32_t)dim1) << 16),       // tensor_dim1 (bits 95:80)
                   (int)(((uint32_t)QN)   << 16),       // tile_dim0 = 300 (bits 127:112)
                   (int)TILE_J,                         // tile_dim1 = 32 (bits 143:128)
                   (int)QN,                             // tensor_dim0_stride = 300
                   0, 0 };                              // dim1_stride unused (2-D tile)
    tdm_v4i gz = { 0, 0, 0, 0 };
#if defined(__clang_major__) && (__clang_major__ >= 23)
    tdm_v8i gz8 = { 0, 0, 0, 0, 0, 0, 0, 0 };
    __builtin_amdgcn_tensor_load_to_lds(g0, g1, gz, gz, gz8, 0);
#else
    __builtin_amdgcn_tensor_load_to_lds(g0, g1, gz, gz, 0);
#endif
    __builtin_amdgcn_s_wait_tensorcnt(0);
  }
  __syncthreads();

  // ---- fused cross-entropy: this block owns rows q = j0..j0+31 of pred_cls ----
  {
    const int w    = t >> 5;            // wave id 0..7
    const int lane = t & 31;
    float wsum = 0.0f;
#pragma unroll
    for (int r4 = 0; r4 < 4; ++r4) {
      const int r = w * 4 + r4;
      const int q = j0 + r;
      if (q < QN) {
        const int gc = gt_cls[(size_t)b * QN + q];
        float xs[10];
        float mx = -3.0e38f;
        float xg = 0.0f;
#pragma unroll
        for (int m = 0; m < 10; ++m) {
          const int col = lane + 32 * m;
          const float v = (col < QN) ? tile[r * QN + col] : -3.0e38f;
          xs[m] = v;
          mx = fmaxf(mx, v);
          if (col == gc) xg = v;
        }
        mx = wave_max32(mx);
        float se = 0.0f;
#pragma unroll
        for (int m = 0; m < 10; ++m) {
          const int col = lane + 32 * m;
          if (col < QN) se += expf(xs[m] - mx);
        }
        se = wave_sum32(se);
        xg = wave_sum32(xg);
        if (lane == 0) {
          const float ce = logf(se) + mx - xg;
          const float wt = gt_validity[(size_t)b * QN + q] ? 1.0f : 0.1f;
          wsum += ce * wt;
        }
      }
    }
    if ((t & 31) == 0) sh_ce[t >> 5] = wsum;
  }
  __syncthreads();
  if (t == 0) {
    float s = 0.0f;
#pragma unroll
    for (int i = 0; i < 8; ++i) s += sh_ce[i];   // fixed order -> deterministic
    ws[(size_t)b * JT + jt] = s;
  }

  // ---- cost tile: cost[b, i, j] for all i, j in [j0, j0+32) ----
  const int ti = t >> 3;                // 0..31 : i offset within 32-row strip
  const int c  = t & 7;                 // j = j0 + c + 8k  (bank-conflict-free LDS reads)

  float4 pb[4];
  float  a1[4];
  bool   jok[4];
#pragma unroll
  for (int k = 0; k < 4; ++k) {
    const int j = j0 + c + 8 * k;
    jok[k] = (j < QN);
    if (jok[k]) {
      pb[k] = *reinterpret_cast<const float4*>(pred_boxes + ((size_t)b * QN + j) * 4);
    } else {
      pb[k] = make_float4(0.f, 0.f, 0.f, 0.f);
    }
    a1[k] = (pb[k].z - pb[k].x) * (pb[k].w - pb[k].y);
  }

  for (int ib = 0; ib < 10; ++ib) {
    const int i = ib * 32 + ti;
    if (i >= QN) continue;
    const float4 gb = *reinterpret_cast<const float4*>(gt_boxes + ((size_t)b * QN + i) * 4);
    const float a2  = (gb.z - gb.x) * (gb.w - gb.y);
    const bool  gv  = gt_validity[(size_t)b * QN + i] != 0;
    float* orow = cost + ((size_t)b * QN + i) * QN;
#pragma unroll
    for (int k = 0; k < 4; ++k) {
      if (!jok[k]) continue;
      const int   j   = j0 + c + 8 * k;
      const float cls = tile[(c + 8 * k) * QN + i];

      const float x1i = fmaxf(pb[k].x, gb.x);
      const float y1i = fmaxf(pb[k].y, gb.y);
      const float x2i = fminf(pb[k].z, gb.z);
      const float y2i = fminf(pb[k].w, gb.w);
      const float inter = fmaxf(x2i - x1i, 0.0f) * fmaxf(y2i - y1i, 0.0f);
      const float uni   = a1[k] + a2 - inter;
      const float iou   = inter / (uni + EPSF);
      const float xc1 = fminf(pb[k].x, gb.x);
      const float yc1 = fminf(pb[k].y, gb.y);
      const float xc2 = fmaxf(pb[k].z, gb.z);
      const float yc2 = fmaxf(pb[k].w, gb.w);
      const float areac = (xc2 - xc1) * (yc2 - yc1);
      const float giou  = iou - (areac - uni) / (areac + EPSF);
      const float l1 = 0.25f * (fabsf(pb[k].x - gb.x) + fabsf(pb[k].y - gb.y) +
                                fabsf(pb[k].z - gb.z) + fabsf(pb[k].w - gb.w));
      const float val = (1.0f - giou) + l1 - cls;
      orow[j] = gv ? val : 0.0f;
    }
  }
}

__global__ __launch_bounds__(256) void detr_ce_reduce_kernel(
    const float* __restrict__ ws, int n, float* __restrict__ out, float scale)
{
  __shared__ float s[256];
  float acc = 0.0f;
  for (int i = threadIdx.x; i < n; i += 256) acc += ws[i];
  s[threadIdx.x] = acc;
  __syncthreads();
#pragma unroll
  for (int off = 128; off > 0; off >>= 1) {
    if ((int)threadIdx.x < off) s[threadIdx.x] += s[threadIdx.x + off];
    __syncthreads();
  }
  if (threadIdx.x == 0) out[0] = s[0] * scale;
}

extern "C" void kernel_launch(void* const* d_in, const int* in_sizes, int n_in,
                              void* d_out, int out_size, void* d_ws, size_t ws_size,
                              hipStream_t stream) {
  (void)in_sizes; (void)n_in; (void)ws_size;
  const float*         pred_boxes  = (const float*)d_in[0];
  const float*         pred_cls    = (const float*)d_in[1];
  const float*         gt_boxes    = (const float*)d_in[2];
  const int*           gt_cls      = (const int*)d_in[3];
  const unsigned char* gt_validity = (const unsigned char*)d_in[4];
  float* out = (float*)d_out;
  float* ws  = (float*)d_ws;

  dim3 grid(JT, BN, 1);
  detr_cost_ce_kernel<<<grid, 256, 0, stream>>>(pred_boxes, pred_cls, gt_boxes,
                                                gt_cls, gt_validity, out, ws);
  detr_ce_reduce_kernel<<<1, 256, 0, stream>>>(ws, BN * JT,
                                               out + ((size_t)out_size - 1),
                                               1.0f / (float)(BN * QN));
}